// WireframeDetector_2748779070290
// MI455X (gfx1250) — compile-verified
//
#include <hip/hip_runtime.h>
#include <cstdint>

// WireframeDetector post-processing for MI455X (gfx1250).
// H=W=2048, C=4, K=out_size/7 (=300).
// ws layout (needs ~17.4 MB):
//   [0, 16MB)        : jloc heatmap (H*W floats) -- L2-resident across passes
//   [+0, 64KB)       : 16384-bin histogram of score float-bits>>16
//   [+, 256B)        : counters: [0]=candidate count, [1]=threshold bits
//   [+, 512KB)       : candidate keys (u64: scoreBits<<32 | ~index)

#define HH 2048
#define WW 2048
#define HWP (HH * WW)
#define NBINS 16384
#define MAXCAND 65536

#define TILE_W 64
#define TILE_H 16
#define LDS_W (TILE_W + 2)
#define LDS_H (TILE_H + 2)

#define WS_JLOC 0
#define WS_HIST (HWP * 4)
#define WS_CNT (WS_HIST + NBINS * 4)
#define WS_CAND (WS_CNT + 256)

typedef unsigned long long ull;
typedef unsigned u32x4 __attribute__((ext_vector_type(4)));
typedef unsigned u32x8 __attribute__((ext_vector_type(8)));

#define NEG_BIG (-3.402823466e+38f)

__device__ __forceinline__ float jloc_of(float x0, float x1, float x2, float x3) {
  float m = fmaxf(fmaxf(x0, x1), fmaxf(x2, x3));
  float e0 = __expf(x0 - m);
  float s = e0 + __expf(x1 - m) + __expf(x2 - m) + __expf(x3 - m);
  return 1.0f - e0 / s;
}

// Pass 1: softmax -> junction-ness map; also zero histogram.
__global__ void jloc_kernel(const float* __restrict__ jl,
                            float* __restrict__ jloc,
                            unsigned* __restrict__ hist) {
  int gid = blockIdx.x * blockDim.x + threadIdx.x;  // one thread = 4 pixels
  if (gid < NBINS) hist[gid] = 0u;

  const float4* c0 = (const float4*)(jl + 0 * HWP);
  const float4* c1 = (const float4*)(jl + 1 * HWP);
  const float4* c2 = (const float4*)(jl + 2 * HWP);
  const float4* c3 = (const float4*)(jl + 3 * HWP);

  // stream-ahead hint (gfx1250 global_prefetch_b8)
  __builtin_prefetch((const char*)(c0 + gid) + 4096, 0, 1);

  float4 a = c0[gid];
  float4 b = c1[gid];
  float4 c = c2[gid];
  float4 d = c3[gid];
  float4 r;
  r.x = jloc_of(a.x, b.x, c.x, d.x);
  r.y = jloc_of(a.y, b.y, c.y, d.y);
  r.z = jloc_of(a.z, b.z, c.z, d.z);
  r.w = jloc_of(a.w, b.w, c.w, d.w);
  ((float4*)jloc)[gid] = r;
}

// Stage a (TILE_H+2)x(TILE_W+2) halo tile of the jloc map into LDS:
//  - interior tiles: one Tensor-Data-Mover descriptor (TENSORcnt path)
//  - border tiles:   per-element async global->LDS loads with -inf halo fill
// then 3x3 NMS.
// PASS 0: histogram score bit-patterns.  PASS 1: append candidates >= threshold.
template <int PASS>
__global__ void nms_kernel(const float* __restrict__ jloc,
                           unsigned* __restrict__ hist,
                           unsigned* __restrict__ cnt,
                           ull* __restrict__ cand) {
  __shared__ float tile[LDS_H * LDS_W];

  const int bx = blockIdx.x % (WW / TILE_W);
  const int by = blockIdx.x / (WW / TILE_W);
  const int x0 = bx * TILE_W - 1;
  const int y0 = by * TILE_H - 1;

  // Low 32 bits of the flat shared pointer == LDS byte offset (gfx1250 aperture).
  const unsigned ldsBase = (unsigned)(uintptr_t)(void*)tile;
  const unsigned long long gbase = (unsigned long long)(uintptr_t)jloc;

  const bool interior =
      (bx > 0) & (bx < (WW / TILE_W) - 1) & (by > 0) & (by < (HH / TILE_H) - 1);

  if (interior) {
    // Whole halo tile is in-bounds: one TDM 2D tile descriptor.
    if (threadIdx.x == 0) {
      unsigned long long ga =
          gbase + (unsigned long long)(unsigned)((y0 * WW + x0) * 4);
      u32x4 g0;
      g0[0] = 1u;                       // count=1, is_restore=0, gather_mode=0
      g0[1] = ldsBase;                  // lds_addr (bytes)
      g0[2] = (unsigned)ga;             // global_addr[31:0]
      g0[3] = ((unsigned)(ga >> 32) & 0x01FFFFFFu) | (2u << 30);  // addr hi | type=2
      u32x8 g1;
      g1[0] = 2u << 16;                 // wg_mask=0, data_size=2 (4 bytes)
      g1[1] = (unsigned)WW << 16;       // tensor_dim0[15:0]   (bits 63:48)
      g1[2] = (unsigned)HH << 16;       // tensor_dim1[15:0]   (bits 95:80)
      g1[3] = (unsigned)LDS_W << 16;    // tile_dim0 = 66      (bits 127:112)
      g1[4] = (unsigned)LDS_H;          // tile_dim1 = 18      (bits 143:128)
      g1[5] = (unsigned)WW;             // tensor_dim0_stride[31:0] = 2048
      g1[6] = 0u;                       // stride hi, tensor_dim1_stride lo
      g1[7] = 0u;
      asm volatile("tensor_load_to_lds %0, %1" : : "s"(g0), "s"(g1) : "memory");
#if defined(__has_builtin) && __has_builtin(__builtin_amdgcn_s_wait_tensorcnt)
      __builtin_amdgcn_s_wait_tensorcnt(0);
#else
      asm volatile("s_wait_tensorcnt 0x0" ::: "memory");
#endif
    }
  } else {
    for (int i = threadIdx.x; i < LDS_H * LDS_W; i += blockDim.x) {
      int ty = i / LDS_W, tx = i - ty * LDS_W;
      int gy = y0 + ty, gx = x0 + tx;
      if ((unsigned)gy < (unsigned)HH && (unsigned)gx < (unsigned)WW) {
        unsigned byteOff = (unsigned)(gy * WW + gx) * 4u;
        unsigned ldsAddr = ldsBase + (unsigned)i * 4u;
        // GVS form: copies 4 bytes from the global base in saddr plus the offset in vaddr to the LDS address in vdst; tracked with ASYNCcnt.
        asm volatile("global_load_async_to_lds_b32 %0, %1, %2"
                     :
                     : "v"(ldsAddr), "v"(byteOff), "s"(gbase)
                     : "memory");
      } else {
        tile[i] = NEG_BIG;  // -inf padding: OOB neighbors never suppress
      }
    }
    asm volatile("s_wait_asynccnt 0x0" ::: "memory");
  }
  __syncthreads();

  const unsigned thr = (PASS == 1) ? cnt[1] : 0u;

  for (int s = 0; s < (TILE_W * TILE_H) / 256; ++s) {
    int lid = (int)threadIdx.x + s * 256;
    int lx = lid % TILE_W, ly = lid / TILE_W;
    int cx = lx + 1, cy = ly + 1;
    float c = tile[cy * LDS_W + cx];
    float n = NEG_BIG;
#pragma unroll
    for (int dy = -1; dy <= 1; ++dy)
#pragma unroll
      for (int dx = -1; dx <= 1; ++dx)
        if (dx != 0 || dy != 0) n = fmaxf(n, tile[(cy + dy) * LDS_W + (cx + dx)]);
    // keep local maxima; jloc in (0,1) so kept scores are > 0
    if (c > 0.0f && c >= n) {
      unsigned bits = __float_as_uint(c);
      if (PASS == 0) {
        atomicAdd(&hist[bits >> 16], 1u);
      } else if (bits >= thr) {
        unsigned pos = atomicAdd(&cnt[0], 1u);
        if (pos < MAXCAND) {
          unsigned idx = (unsigned)((y0 + cy) * WW + (x0 + cx));
          cand[pos] = ((ull)bits << 32) | (ull)(unsigned)(~idx);
        }
      }
    }
  }
}

// Find largest bin B s.t. count(bins >= B) >= K; threshold = B<<16.
__global__ void thresh_kernel(const unsigned* __restrict__ hist,
                              unsigned* __restrict__ cnt, int K) {
  if (threadIdx.x == 0) {
    unsigned acc = 0, thr = 0;
    for (int b = NBINS - 1; b >= 0; --b) {
      acc += hist[b];
      if ((int)acc >= K) { thr = (unsigned)b << 16; break; }
    }
    cnt[1] = thr;
    cnt[0] = 0;  // reset candidate counter for the append pass
  }
}

// Single block: iteratively extract K max keys, gather offsets/logits.
__global__ void select_kernel(const float* __restrict__ jl,
                              const float* __restrict__ jo,
                              ull* __restrict__ cand,
                              const unsigned* __restrict__ cnt,
                              float* __restrict__ out, int K) {
  __shared__ ull skey[256];
  __shared__ int spos[256];
  const int tid = (int)threadIdx.x;
  unsigned c0 = cnt[0];
  int n = (int)(c0 < (unsigned)MAXCAND ? c0 : (unsigned)MAXCAND);

  for (int k = 0; k < K; ++k) {
    ull best = 0;
    int bpos = -1;
    for (int i = tid; i < n; i += 256) {
      ull v = cand[i];
      if (v > best) { best = v; bpos = i; }
    }
    skey[tid] = best;
    spos[tid] = bpos;
    __syncthreads();
#pragma unroll
    for (int off = 128; off > 0; off >>= 1) {
      if (tid < off && skey[tid + off] > skey[tid]) {
        skey[tid] = skey[tid + off];
        spos[tid] = spos[tid + off];
      }
      __syncthreads();
    }
    if (tid == 0) {
      ull b = skey[0];
      float row[7] = {0.f, 0.f, 0.f, 0.f, 0.f, 0.f, 0.f};
      if (b != 0) {
        cand[spos[0]] = 0;  // remove winner
        unsigned bits = (unsigned)(b >> 32);
        unsigned idx = ~(unsigned)(b & 0xFFFFFFFFull);
        float ox = 1.0f / (1.0f + __expf(-jo[idx]));          // sigmoid-0.5+0.5
        float oy = 1.0f / (1.0f + __expf(-jo[HWP + idx]));
        row[0] = (float)(idx % WW) + ox;
        row[1] = (float)(idx / WW) + oy;
        row[2] = __uint_as_float(bits);
        row[3] = jl[0 * HWP + idx];
        row[4] = jl[1 * HWP + idx];
        row[5] = jl[2 * HWP + idx];
        row[6] = jl[3 * HWP + idx];
      }
#pragma unroll
      for (int j = 0; j < 7; ++j) out[k * 7 + j] = row[j];
    }
    __threadfence_block();
    __syncthreads();
  }
}

extern "C" void kernel_launch(void* const* d_in, const int* in_sizes, int n_in,
                              void* d_out, int out_size, void* d_ws, size_t ws_size,
                              hipStream_t stream) {
  (void)in_sizes; (void)n_in; (void)ws_size;
  const float* jl = (const float*)d_in[0];   // [1,4,H,W]
  const float* jo = (const float*)d_in[1];   // [1,2,H,W]
  const int K = out_size / 7;                // == topk (300)

  char* ws = (char*)d_ws;
  float* jloc = (float*)(ws + WS_JLOC);
  unsigned* hist = (unsigned*)(ws + WS_HIST);
  unsigned* cnt = (unsigned*)(ws + WS_CNT);
  ull* cand = (ull*)(ws + WS_CAND);
  float* out = (float*)d_out;

  const int nmsBlocks = (WW / TILE_W) * (HH / TILE_H);  // 32*128 = 4096

  jloc_kernel<<<HWP / 4 / 256, 256, 0, stream>>>(jl, jloc, hist);
  nms_kernel<0><<<nmsBlocks, 256, 0, stream>>>(jloc, hist, cnt, cand);
  thresh_kernel<<<1, 32, 0, stream>>>(hist, cnt, K);
  nms_kernel<1><<<nmsBlocks, 256, 0, stream>>>(jloc, hist, cnt, cand);
  select_kernel<<<1, 256, 0, stream>>>(jl, jo, cand, cnt, out, K);
}